// IntraListDiversityScore_19378892440031
// MI455X (gfx1250) — compile-verified
//
#include <hip/hip_runtime.h>
#include <hip/hip_bf16.h>
#include <math.h>

// ---------------------------------------------------------------------------
// IntraListDiversityScore for MI455X (gfx1250, wave32)
//
// Kernel 1: one 128-thread block per user. Loads the 50 item ids to LDS,
//           decodes the 1225 (i<j) pairs with a BRANCHLESS closed-form
//           triangular decode (straight-line body so the compiler clusters
//           the 9 independent gathers for max memory-level parallelism),
//           wave32 shfl reduction, one f32 partial per user into d_ws.
// Kernel 2: single wave32. Reduces the 16384 partials with
//           V_WMMA_F32_16X16X4_F32 (B = ones): sum over all elements of the
//           accumulator equals 16 * (sum of everything fed through A),
//           independent of the WMMA lane/VGPR striping.
// ---------------------------------------------------------------------------

#define K_ITEMS 50
#define N_PAIRS 1225          // K*(K-1)/2
#define BLK 128
#define FULL_ITERS 9          // t=0..8: p = tid + t*128 <= 1151 < 1225 always

typedef __attribute__((ext_vector_type(2))) float v2f;
typedef __attribute__((ext_vector_type(8))) float v8f;

// Branchless decode of linear upper-triangular pair index p -> (i, j), i < j,
// for K = 50. Row start off(i) = i*(99-i)/2. At row starts the discriminant
// 9801 - 8*off(i) = (99-2i)^2 is a perfect square, so sqrtf is exact there;
// elsewhere the fractional part is >= ~0.02 from an integer while FP error is
// <= ~1e-3, so the floor estimate is off by at most one. Two select-based
// fixups (no loops, no branches) make it exact.
__device__ __forceinline__ void decode_pair(int p, int& i_out, int& j_out) {
    int i = (int)((99.0f - sqrtf((float)(9801 - 8 * p))) * 0.5f);
    i = i < 0 ? 0 : (i > 48 ? 48 : i);
    // step up if next row start <= p      (off(i+1) = (i+1)*(98-i)/2)
    const int off_next = ((i + 1) * (98 - i)) >> 1;
    i += (off_next <= p) ? 1 : 0;
    // step down if current row start > p  (off(i) = i*(99-i)/2)
    const int off_cur = (i * (99 - i)) >> 1;
    i -= (off_cur > p) ? 1 : 0;
    const int off = (i * (99 - i)) >> 1;
    i_out = i;
    j_out = p - off + i + 1;
}

__global__ __launch_bounds__(BLK) void ild_pair_gather(
    const int* __restrict__ rec,      // [B, 50] int32
    const float* __restrict__ dist,   // [n_items, n_items] f32
    float* __restrict__ partial,      // [B] f32 (raw pair sums)
    int n_items) {
    __shared__ int s_rec[K_ITEMS];
    __shared__ float s_wave[BLK / 32];

    const int user = blockIdx.x;
    const int tid  = threadIdx.x;

    if (tid < K_ITEMS)
        s_rec[tid] = rec[(size_t)user * K_ITEMS + tid];
    __syncthreads();

    float acc = 0.0f;

    // 9 always-in-range pairs: fully straight-line, loads issued back-to-back.
#pragma unroll
    for (int t = 0; t < FULL_ITERS; ++t) {
        int i, j;
        decode_pair(tid + t * BLK, i, j);
        const size_t idx = (size_t)s_rec[i] * (size_t)n_items + (size_t)s_rec[j];
        acc += dist[idx];
    }

    // guarded tail (t = 9): valid only for tid <= 72
    {
        const int p = tid + FULL_ITERS * BLK;
        if (p < N_PAIRS) {
            int i, j;
            decode_pair(p, i, j);
            const size_t idx = (size_t)s_rec[i] * (size_t)n_items + (size_t)s_rec[j];
            acc += dist[idx];
        }
    }

    // wave32 butterfly reduction
#pragma unroll
    for (int off = 16; off > 0; off >>= 1)
        acc += __shfl_xor(acc, off, 32);

    if ((tid & 31) == 0) s_wave[tid >> 5] = acc;
    __syncthreads();

    if (tid == 0) {
        float s = 0.0f;
#pragma unroll
        for (int w = 0; w < BLK / 32; ++w) s += s_wave[w];
        partial[user] = s;
    }
}

// Single-wave WMMA reduction: n must be a multiple of 64 (16384 is).
// D = A x B + C with B = all ones:
//   sum_all(D) = 16 * sum_all(A) + sum_all(C)   (layout-invariant)
__global__ __launch_bounds__(32) void ild_wmma_reduce(
    const float* __restrict__ partial,
    float* __restrict__ out,
    int n,            // number of partials (multiple of 64)
    float inv_scale)  // 1 / (16 * K*(K-1) * B)
{
    const int lane = threadIdx.x;  // 0..31, full wave, EXEC all ones

    v8f c = {};
    v2f ones;
    ones.x = 1.0f;
    ones.y = 1.0f;

    for (int base = 0; base < n; base += 64) {
        v2f a;
        a.x = partial[base + lane];
        a.y = partial[base + 32 + lane];
        // (neg_a, A, neg_b, B, c_mod, C, reuse_a, reuse_b)
        c = __builtin_amdgcn_wmma_f32_16x16x4_f32(
            false, a, false, ones, (short)0, c, false, false);
    }

    float s = c[0] + c[1] + c[2] + c[3] + c[4] + c[5] + c[6] + c[7];
#pragma unroll
    for (int off = 16; off > 0; off >>= 1)
        s += __shfl_xor(s, off, 32);

    if (lane == 0)
        out[0] = s * inv_scale;
}

extern "C" void kernel_launch(void* const* d_in, const int* in_sizes, int n_in,
                              void* d_out, int out_size, void* d_ws, size_t ws_size,
                              hipStream_t stream) {
    const int*   rec  = (const int*)d_in[0];      // [B, 50] int32
    const float* dist = (const float*)d_in[1];    // [n_items, n_items] f32
    float* out        = (float*)d_out;            // scalar f32
    float* partial    = (float*)d_ws;             // B floats of scratch

    const int B = in_sizes[0] / K_ITEMS;                        // 16384
    const int n_items = (int)(sqrt((double)in_sizes[1]) + 0.5); // 20000

    ild_pair_gather<<<B, BLK, 0, stream>>>(rec, dist, partial, n_items);

    const float inv_scale =
        1.0f / (16.0f * (float)(K_ITEMS * (K_ITEMS - 1)) * (float)B);
    ild_wmma_reduce<<<1, 32, 0, stream>>>(partial, out, B, inv_scale);
}